// FreLoss_5643587027145
// MI455X (gfx1250) — compile-verified
//
#include <hip/hip_runtime.h>
#include <hip/hip_bf16.h>
#include <math.h>

typedef __attribute__((ext_vector_type(2))) float v2f;
typedef __attribute__((ext_vector_type(8))) float v8f;

#define NLAT 128
#define NLON 256
#define LMAX 50
#define MMAX 50
#define NPTS 2048
#define NGRID (NLAT * NLON)   // 32768
#define MPAD 64               // padded m / l dimension
#define PI_F 3.14159265358979323846f

// ---------------- workspace layout (floats) ----------------
// sph   : 4*2048*2   = 16384
// r     : 4*2048     = 8192
// field : 4*128*256  = 131072
// Ccos  : 256*64     = 16384
// PB    : 50*128*64  = 409600   (PCTW * quadrature, [m][k][l])
// XR    : 512*64     = 32768
// coef  : 4*64*64    = 16384
#define OFF_SPH   0
#define OFF_R     16384
#define OFF_FIELD 24576
#define OFF_C     155648
#define OFF_PB    172032
#define OFF_XR    581632
#define OFF_COEF  614400

// ---------------- 1) spherical coords ----------------
__global__ void sph_kernel(const float* __restrict__ pred,
                           const float* __restrict__ target,
                           float* __restrict__ sph, float* __restrict__ rr) {
    int idx = blockIdx.x * blockDim.x + threadIdx.x;
    if (idx >= 4 * NPTS) return;
    int cloud = idx / NPTS;          // 0,1 = pred batches; 2,3 = target batches
    int p = idx - cloud * NPTS;
    const float* src = (cloud < 2) ? pred : target;
    int off = ((cloud & 1) * NPTS + p) * 3;
    float x = src[off], y = src[off + 1], z = src[off + 2];
    float r = sqrtf(x * x + y * y + z * z);
    float theta = acosf(x / r);
    float ryz = sqrtf(y * y + z * z);
    float a = acosf(y / ryz);
    float phi = (z < 0.0f) ? (2.0f * PI_F - a) : a;
    phi -= PI_F;                     // reference shifts azimuth by -pi
    sph[idx * 2 + 0] = theta;
    sph[idx * 2 + 1] = phi;
    rr[idx] = r;
}

// ---------------- 2) cos DFT matrix: C[j][m] = cos(2pi m j/256)*(2pi/256) ----------------
__global__ void cos_kernel(float* __restrict__ C) {
    int idx = blockIdx.x * blockDim.x + threadIdx.x;
    if (idx >= NLON * MPAD) return;
    int j = idx / MPAD, m = idx - j * MPAD;
    float v = 0.0f;
    if (m < MMAX) {
        float ang = (2.0f * PI_F) * (float)(m * j) / (float)NLON;
        v = cosf(ang) * (2.0f * PI_F / (float)NLON);
    }
    C[idx] = v;
}

// ---------------- 3) Legendre * CC-quadrature: PB[m][k][l] ----------------
__global__ void leg_kernel(float* __restrict__ PB) {
    int idx = blockIdx.x * blockDim.x + threadIdx.x;
    if (idx >= MMAX * NLAT) return;
    int m = idx / NLAT, k = idx - m * NLAT;
    const int Nn = NLAT - 1;                 // 127
    float theta = PI_F * (float)k / (float)Nn;
    float cost = cosf(theta);
    float sint = sqrtf(fmaxf(1.0f - cost * cost, 0.0f));
    // Clenshaw-Curtis weight (Nn odd -> b==2 always)
    float v = 0.0f;
    for (int kk = 1; kk <= Nn / 2; ++kk)
        v += 2.0f * cosf(2.0f * (float)kk * theta) / (float)(4 * kk * kk - 1);
    float w = (2.0f / (float)Nn) * (1.0f - v);
    if (k == 0 || k == Nn) w *= 0.5f;
    // pmm
    float pmm = sqrtf(1.0f / (4.0f * PI_F));
    for (int mm = 1; mm <= m; ++mm)
        pmm = -pmm * sqrtf((2.0f * mm + 1.0f) / (2.0f * mm)) * sint;
    float pm2 = 0.0f, pm1 = 0.0f;
    float* out = PB + ((size_t)m * NLAT + k) * MPAD;
    for (int l = 0; l < MPAD; ++l) {
        float val;
        if (l < m || l >= LMAX) val = 0.0f;
        else if (l == m)        val = pmm;
        else if (l == m + 1)    val = sqrtf(2.0f * m + 3.0f) * cost * pmm;
        else {
            float fl = (float)l;
            float aa = sqrtf((4.0f * fl * fl - 1.0f) / (fl * fl - (float)(m * m)));
            float bb = sqrtf(((2.0f * fl + 1.0f) * (fl - 1.0f + m) * (fl - 1.0f - m)) /
                             ((2.0f * fl - 3.0f) * (fl * fl - (float)(m * m))));
            val = aa * cost * pm1 - bb * pm2;
        }
        out[l] = val * w;
        pm2 = pm1; pm1 = val;
    }
}

// ---------------- 4) brute-force 3-NN + dist-weighted interp (LDS staging) ----------------
__global__ void knn_kernel(const float* __restrict__ sph, const float* __restrict__ rr,
                           float* __restrict__ field) {
    __shared__ float4 pts[NPTS];             // 32 KB of the 320 KB WGP LDS
    int cloud = blockIdx.x >> 7;             // 128 blocks of 256 grid pts per cloud
    int gbase = (blockIdx.x & 127) * 256;
    for (int p = threadIdx.x; p < NPTS; p += 256) {
        float sx = sph[(cloud * NPTS + p) * 2 + 0];
        float sy = sph[(cloud * NPTS + p) * 2 + 1];
        float rv = rr[cloud * NPTS + p];
        pts[p] = make_float4(sx, sy, rv, sx * sx + sy * sy);
    }
    __syncthreads();
    int g = gbase + threadIdx.x;             // 0..32767
    float gx = (float)(g >> 8) * (PI_F / (float)NLAT);
    float gy = (float)((g & 255) - NLAT) * (PI_F / (float)NLAT);
    float g2 = gx * gx + gy * gy;
    float d0 = 1e30f, d1 = 1e30f, d2 = 1e30f;
    float ax0 = 0, ay0 = 0, ar0 = 0, ax1 = 0, ay1 = 0, ar1 = 0, ax2 = 0, ay2 = 0, ar2 = 0;
    for (int p = 0; p < NPTS; ++p) {
        float4 q = pts[p];
        // expanded-form distance (matches reference's selection metric)
        float d = g2 + q.w - 2.0f * (gx * q.x + gy * q.y);
        if (d < d2) {
            if (d < d1) {
                d2 = d1; ax2 = ax1; ay2 = ay1; ar2 = ar1;
                if (d < d0) {
                    d1 = d0; ax1 = ax0; ay1 = ay0; ar1 = ar0;
                    d0 = d;  ax0 = q.x; ay0 = q.y; ar0 = q.z;
                } else {
                    d1 = d;  ax1 = q.x; ay1 = q.y; ar1 = q.z;
                }
            } else {
                d2 = d; ax2 = q.x; ay2 = q.y; ar2 = q.z;
            }
        }
    }
    // recompute direct-form sq-dists of selected (matches reference) and weight
    float k0 = (gx - ax0) * (gx - ax0) + (gy - ay0) * (gy - ay0);
    float k1 = (gx - ax1) * (gx - ax1) + (gy - ay1) * (gy - ay1);
    float k2 = (gx - ax2) * (gx - ax2) + (gy - ay2) * (gy - ay2);
    float s = k0 + k1 + k2;
    field[cloud * NGRID + g] = (k0 * ar0 + k1 * ar1 + k2 * ar2) / s;
}

// ---------------- 5) DFT stage: XR(512x64) = field(512x256) * C(256x64), f32 WMMA ----------------
__global__ void gemm_dft_kernel(const float* __restrict__ field,
                                const float* __restrict__ C,
                                float* __restrict__ XR) {
    int tileM = blockIdx.x >> 2;             // 32 tiles of rows
    int tileN = blockIdx.x & 3;              // 4 tiles of cols
    int lane = threadIdx.x;                  // wave32, one wave per block
    int Mrow = tileM * 16 + (lane & 15);
    int Ncol = tileN * 16 + (lane & 15);
    int kb = (lane >> 4) * 2;
    v8f acc = {};
#pragma unroll 8
    for (int k0 = 0; k0 < NLON; k0 += 4) {
        v2f a, b;
        a.x = field[Mrow * NLON + k0 + kb];
        a.y = field[Mrow * NLON + k0 + kb + 1];
        b.x = C[(k0 + kb) * MPAD + Ncol];
        b.y = C[(k0 + kb + 1) * MPAD + Ncol];
        acc = __builtin_amdgcn_wmma_f32_16x16x4_f32(false, a, false, b,
                                                    (short)0, acc, false, false);
    }
#pragma unroll
    for (int v = 0; v < 8; ++v) {
        int row = tileM * 16 + v + 8 * (lane >> 4);
        XR[row * MPAD + Ncol] = acc[v];
    }
}

// ---------------- 6) Legendre stage, batched over m: coef[c][l][m] ----------------
__global__ void gemm_leg_kernel(const float* __restrict__ XR,
                                const float* __restrict__ PB,
                                float* __restrict__ coef) {
    int m = blockIdx.x;                      // 0..49
    int nt = blockIdx.y;                     // 0..3 (l tiles)
    int lane = threadIdx.x;
    int Mrow = lane & 15;                    // cloud (only 0..3 valid)
    int Ncol = nt * 16 + (lane & 15);        // l
    int kb = (lane >> 4) * 2;
    bool mv = (Mrow < 4);
    v8f acc = {};
#pragma unroll 8
    for (int k0 = 0; k0 < NLAT; k0 += 4) {
        v2f a, b;
        a.x = mv ? XR[(Mrow * NLAT + k0 + kb) * MPAD + m] : 0.0f;
        a.y = mv ? XR[(Mrow * NLAT + k0 + kb + 1) * MPAD + m] : 0.0f;
        b.x = PB[((size_t)m * NLAT + k0 + kb) * MPAD + Ncol];
        b.y = PB[((size_t)m * NLAT + k0 + kb + 1) * MPAD + Ncol];
        acc = __builtin_amdgcn_wmma_f32_16x16x4_f32(false, a, false, b,
                                                    (short)0, acc, false, false);
    }
#pragma unroll
    for (int v = 0; v < 8; ++v) {
        int c = v + 8 * (lane >> 4);         // cloud row of D
        if (c < 4) coef[c * (MPAD * MPAD) + Ncol * MPAD + m] = acc[v];
    }
}

// ---------------- 7) final weighted-SSE reduction -> scalar ----------------
__global__ void reduce_kernel(const float* __restrict__ coef, float* __restrict__ out) {
    __shared__ float red[256];
    float acc = 0.0f;
    for (int t = threadIdx.x; t < 2 * LMAX * MMAX; t += 256) {
        int b = t / (LMAX * MMAX);
        int rem = t - b * (LMAX * MMAX);
        int l = rem / MMAX, m = rem - l * MMAX;
        float pc = coef[b * (MPAD * MPAD) + l * MPAD + m];
        float tc = coef[(2 + b) * (MPAD * MPAD) + l * MPAD + m];
        float diff = pc - tc;
        float q = (float)(LMAX - (l + 1));
        float rect = expf(-(q * q) / (2.0f * (float)(LMAX * LMAX)));
        acc += diff * diff * rect;
    }
    red[threadIdx.x] = acc;
    __syncthreads();
    for (int s = 128; s > 0; s >>= 1) {
        if (threadIdx.x < s) red[threadIdx.x] += red[threadIdx.x + s];
        __syncthreads();
    }
    if (threadIdx.x == 0) out[0] = red[0] * 0.5f;   // mean over B=2
}

extern "C" void kernel_launch(void* const* d_in, const int* in_sizes, int n_in,
                              void* d_out, int out_size, void* d_ws, size_t ws_size,
                              hipStream_t stream) {
    const float* pred = (const float*)d_in[0];
    const float* target = (const float*)d_in[1];
    float* ws = (float*)d_ws;
    float* sph   = ws + OFF_SPH;
    float* rr    = ws + OFF_R;
    float* field = ws + OFF_FIELD;
    float* C     = ws + OFF_C;
    float* PB    = ws + OFF_PB;
    float* XR    = ws + OFF_XR;
    float* coef  = ws + OFF_COEF;
    float* out   = (float*)d_out;

    sph_kernel<<<(4 * NPTS + 255) / 256, 256, 0, stream>>>(pred, target, sph, rr);
    cos_kernel<<<(NLON * MPAD + 255) / 256, 256, 0, stream>>>(C);
    leg_kernel<<<(MMAX * NLAT + 255) / 256, 256, 0, stream>>>(PB);
    knn_kernel<<<4 * (NGRID / 256), 256, 0, stream>>>(sph, rr, field);
    gemm_dft_kernel<<<128, 32, 0, stream>>>(field, C, XR);
    gemm_leg_kernel<<<dim3(MMAX, 4), 32, 0, stream>>>(XR, PB, coef);
    reduce_kernel<<<1, 256, 0, stream>>>(coef, out);
}